// PhotoVoModel_82575041233059
// MI455X (gfx1250) — compile-verified
//
#include <hip/hip_runtime.h>
#include <math.h>

// ---------------------------------------------------------------------------
// PhotoVoModel forward for MI455X (gfx1250), fp32 end-to-end using
// V_WMMA_F32_16X16X4_F32 (exact fp32 semantics, matrix pipe).
// ---------------------------------------------------------------------------

typedef __attribute__((ext_vector_type(2))) float v2f;
typedef __attribute__((ext_vector_type(8))) float v8f;

#define D_   384
#define LDF  388      // LDS row stride for 16x384 block (4-bank offset)
#define KC   64
#define LDA  68       // LDS row stride for A-chunk staging
#define BB   16
#define NN   2048     // patches per batch (2*M)
#define MM   1024
#define NKP  4096
#define IH   376
#define IW   1241
#define BN   (BB*NN)  // 32768 rows

__device__ __forceinline__ v8f vzero8() {
  v8f z = {0.f,0.f,0.f,0.f,0.f,0.f,0.f,0.f};
  return z;
}

__device__ __forceinline__ v8f wmma4(v2f a, v2f b, v8f c) {
  // D = A(16x4, f32) x B(4x16, f32) + C(16x16, f32)
  return __builtin_amdgcn_wmma_f32_16x16x4_f32(false, a, false, b,
                                               (short)0, c, false, false);
}

// One wave accumulates 3 tiles (cols (wave+8*t)*16 .. +15) over k in
// [kbeg,kend) of an A block staged in LDS (row stride lda).  B is a global
// KxD_ row-major weight; kglob is the global-K offset of the LDS chunk.
__device__ __forceinline__ void gemm_acc3(const float* Alds, int lda,
                                          int kbeg, int kend,
                                          const float* __restrict__ Bg,
                                          int kglob, v8f acc[3],
                                          int wave, int lane_lo, int hi) {
  for (int kk = kbeg; kk < kend; kk += 4) {
    v2f a;
    a.x = Alds[lane_lo*lda + kk + 2*hi];       // A[row][k], A[row][k+1]
    a.y = Alds[lane_lo*lda + kk + 2*hi + 1];   // lanes 16-31: K offset +2
    int kr = kglob + kk + 2*hi;
    const float* b0 = Bg + (size_t)kr * D_;
#pragma unroll
    for (int t = 0; t < 3; ++t) {
      int col = (wave + 8*t)*16 + lane_lo;
      v2f b;
      b.x = b0[col];
      b.y = b0[D_ + col];
      acc[t] = wmma4(a, b, acc[t]);
    }
  }
}

// LayerNorm over a 16x384 LDS block; 256 threads = 16 rows x 16 segments.
__device__ __forceinline__ void layernorm16(float (*F)[LDF],
                                            const float* __restrict__ g,
                                            const float* __restrict__ bta,
                                            int tid) {
  int row = tid >> 4, seg = tid & 15;
  float s = 0.f;
#pragma unroll
  for (int j = 0; j < 24; ++j) s += F[row][seg + 16*j];
#pragma unroll
  for (int m = 8; m >= 1; m >>= 1) s += __shfl_xor(s, m, 16);
  float mu = s * (1.0f/384.0f);
  float s2 = 0.f;
#pragma unroll
  for (int j = 0; j < 24; ++j) { float d = F[row][seg + 16*j] - mu; s2 += d*d; }
#pragma unroll
  for (int m = 8; m >= 1; m >>= 1) s2 += __shfl_xor(s2, m, 16);
  float rsq = rsqrtf(s2 * (1.0f/384.0f) + 1e-5f);
#pragma unroll
  for (int j = 0; j < 24; ++j) {
    int col = seg + 16*j;
    F[row][col] = (F[row][col] - mu) * rsq * g[col] + bta[col];
  }
}

// ---------------------------------------------------------------------------
// K0: gather per-row coords / scores from matches.
// ---------------------------------------------------------------------------
__global__ __launch_bounds__(256) void prep_kernel(
    const int* __restrict__ matches,
    const float* __restrict__ kpts0, const float* __restrict__ kpts1,
    const float* __restrict__ scores0, const float* __restrict__ scores1,
    float* __restrict__ coords, float* __restrict__ sc) {
  int idx = blockIdx.x * 256 + threadIdx.x;
  if (idx >= BN) return;
  int b = idx >> 11, n = idx & (NN - 1);
  float cx, cy, s;
  if (n < MM) {
    int m0 = matches[((size_t)b*MM + n)*2 + 0];
    int m1 = matches[((size_t)b*MM + n)*2 + 1];
    bool valid = m1 > -1;
    int i0 = valid ? m0 : 0;
    const float* kp = kpts0 + ((size_t)b*NKP + i0)*2;
    cx = valid ? kp[0] : -1.0f;
    cy = valid ? kp[1] : -1.0f;
    s  = valid ? scores0[(size_t)b*NKP + i0] : 0.0f;
  } else {
    int j = n - MM;
    int m1 = matches[((size_t)b*MM + j)*2 + 1];
    bool valid = m1 > -1;
    int i1 = valid ? m1 : 0;
    const float* kp = kpts1 + ((size_t)b*NKP + i1)*2;
    cx = valid ? kp[0] : -1.0f;
    cy = valid ? kp[1] : -1.0f;
    s  = valid ? scores1[(size_t)b*NKP + i1] : 0.0f;
  }
  coords[(size_t)idx*2 + 0] = cx;
  coords[(size_t)idx*2 + 1] = cy;
  sc[idx] = s;
}

// ---------------------------------------------------------------------------
// K1: fused per-16-row-block pipeline:
//   acc1 = gather(patch)@patch_proj_w + b  -> LN1 -> F (LDS)
//   F += relu(coords@pos_w1+b1)@pos_w2 + pos_b2         (posh built on the fly)
//   acc3 = F@score_w[0:384] + score_b + s*score_w[384]  -> relu -> LN2 -> F
//   Kb = F@wk + bk ; Vb = F@wv + bv
// ---------------------------------------------------------------------------
__global__ __launch_bounds__(256) void fused_patch_kernel(
    const float* __restrict__ im0, const float* __restrict__ im1,
    const float* __restrict__ Kmat,
    const float* __restrict__ ppw, const float* __restrict__ ppb,
    const float* __restrict__ ln1g, const float* __restrict__ ln1b,
    const float* __restrict__ pw1, const float* __restrict__ pb1,
    const float* __restrict__ pw2, const float* __restrict__ pb2,
    const float* __restrict__ scw, const float* __restrict__ scb,
    const float* __restrict__ ln2g, const float* __restrict__ ln2b,
    const float* __restrict__ wk, const float* __restrict__ bk,
    const float* __restrict__ wv, const float* __restrict__ bv,
    const float* __restrict__ coords, const float* __restrict__ sc,
    float* __restrict__ Kb, float* __restrict__ Vb) {
  __shared__ float F[16][LDF];
  __shared__ float Ach[16][LDA];
  __shared__ float rc[16][2];
  __shared__ float rs[16];

  int tid = threadIdx.x;
  int wave = tid >> 5, lane = tid & 31, lane_lo = lane & 15, hi = lane >> 4;
  int row0 = blockIdx.x * 16;            // 2048 % 16 == 0 -> single batch
  int bidx = row0 >> 11;

  if (tid < 16) rs[tid] = sc[row0 + tid];
  if (tid < 32) rc[tid >> 1][tid & 1] = coords[(size_t)(row0 + (tid >> 1))*2 + (tid & 1)];
  __syncthreads();

  float fx  = Kmat[bidx*9 + 0], cx0 = Kmat[bidx*9 + 2];
  float fy  = Kmat[bidx*9 + 4], cy0 = Kmat[bidx*9 + 5];

  // ---- Phase 1: gathered patches (K=320) @ patch_proj_w -------------------
  v8f acc[3] = {};
  for (int c = 0; c < 5; ++c) {          // channel c == K-chunk c (64 = 8x8)
    __syncthreads();
    for (int e = tid; e < 16*64; e += 256) {
      int r = e >> 6, kk = e & 63;
      int py = kk >> 3, px = kk & 7;
      int xi = (int)rintf(rc[r][0]); xi = xi < 4 ? 4 : (xi > IW-5 ? IW-5 : xi);
      int yi = (int)rintf(rc[r][1]); yi = yi < 4 ? 4 : (yi > IH-5 ? IH-5 : yi);
      int xa = xi + px - 4, ya = yi + py - 4;
      float val;
      if (c < 3) {
        int n = (row0 + r) & (NN - 1);
        const float* img = (n < MM) ? im0 : im1;
        val = img[(((size_t)bidx*3 + c)*IH + ya)*IW + xa];
      } else if (c == 3) {
        val = ((float)xa - cx0) / fx;     // intrinsics layer u
      } else {
        val = ((float)ya - cy0) / fy;     // intrinsics layer v
      }
      Ach[r][kk] = val;
    }
    __syncthreads();
    gemm_acc3(&Ach[0][0], LDA, 0, 64, ppw, c*64, acc, wave, lane_lo, hi);
  }
  __syncthreads();
#pragma unroll
  for (int t = 0; t < 3; ++t) {
    int col = (wave + 8*t)*16 + lane_lo;
#pragma unroll
    for (int r = 0; r < 8; ++r) F[8*hi + r][col] = acc[t][r] + ppb[col];
  }
  __syncthreads();
  layernorm16(F, ln1g, ln1b, tid);
  __syncthreads();

  // ---- Phase 2: + pos MLP (posh built on the fly, K=384) ------------------
#pragma unroll
  for (int t = 0; t < 3; ++t) acc[t] = vzero8();
  for (int c = 0; c < 6; ++c) {
    __syncthreads();
    for (int e = tid; e < 16*64; e += 256) {
      int r = e >> 6, kk = e & 63;
      int k = c*64 + kk;
      float hv = rc[r][0]*pw1[k] + rc[r][1]*pw1[D_ + k] + pb1[k];
      Ach[r][kk] = hv > 0.f ? hv : 0.f;
    }
    __syncthreads();
    gemm_acc3(&Ach[0][0], LDA, 0, 64, pw2, c*64, acc, wave, lane_lo, hi);
  }
  __syncthreads();
#pragma unroll
  for (int t = 0; t < 3; ++t) {
    int col = (wave + 8*t)*16 + lane_lo;
#pragma unroll
    for (int r = 0; r < 8; ++r) F[8*hi + r][col] += acc[t][r] + pb2[col];
  }
  __syncthreads();

  // ---- Phase 3: score GEMM (K=385 = 384 + rank-1 score row), relu, LN2 ----
#pragma unroll
  for (int t = 0; t < 3; ++t) acc[t] = vzero8();
  gemm_acc3(&F[0][0], LDF, 0, D_, scw, 0, acc, wave, lane_lo, hi);
  float epi[3][8];
#pragma unroll
  for (int t = 0; t < 3; ++t) {
    int col = (wave + 8*t)*16 + lane_lo;
#pragma unroll
    for (int r = 0; r < 8; ++r) {
      int m = 8*hi + r;
      float x = acc[t][r] + scb[col] + rs[m]*scw[(size_t)D_*D_ + col];
      epi[t][r] = x > 0.f ? x : 0.f;
    }
  }
  __syncthreads();                        // all waves done reading F
#pragma unroll
  for (int t = 0; t < 3; ++t) {
    int col = (wave + 8*t)*16 + lane_lo;
#pragma unroll
    for (int r = 0; r < 8; ++r) F[8*hi + r][col] = epi[t][r];
  }
  __syncthreads();
  layernorm16(F, ln2g, ln2b, tid);        // F now holds patch_embs
  __syncthreads();

  // ---- Phase 4: k = F@wk + bk ; v = F@wv + bv (F read-only) ---------------
#pragma unroll
  for (int t = 0; t < 3; ++t) acc[t] = vzero8();
  gemm_acc3(&F[0][0], LDF, 0, D_, wk, 0, acc, wave, lane_lo, hi);
#pragma unroll
  for (int t = 0; t < 3; ++t) {
    int col = (wave + 8*t)*16 + lane_lo;
#pragma unroll
    for (int r = 0; r < 8; ++r)
      Kb[(size_t)(row0 + 8*hi + r)*D_ + col] = acc[t][r] + bk[col];
  }
#pragma unroll
  for (int t = 0; t < 3; ++t) acc[t] = vzero8();
  gemm_acc3(&F[0][0], LDF, 0, D_, wv, 0, acc, wave, lane_lo, hi);
#pragma unroll
  for (int t = 0; t < 3; ++t) {
    int col = (wave + 8*t)*16 + lane_lo;
#pragma unroll
    for (int r = 0; r < 8; ++r)
      Vb[(size_t)(row0 + 8*hi + r)*D_ + col] = acc[t][r] + bv[col];
  }
}

// ---------------------------------------------------------------------------
// K2: q = image_embs @ wq + bq   (single 16-row block, one workgroup)
// ---------------------------------------------------------------------------
__global__ __launch_bounds__(256) void qproj_kernel(
    const float* __restrict__ embs, const float* __restrict__ wq,
    const float* __restrict__ bq, float* __restrict__ qb) {
  __shared__ float A[16][LDF];
  int tid = threadIdx.x;
  for (int e = tid; e < 16*D_; e += 256) A[e / D_][e % D_] = embs[e];
  __syncthreads();
  int wave = tid >> 5, lane = tid & 31, lane_lo = lane & 15, hi = lane >> 4;
  v8f acc[3] = {};
  gemm_acc3(&A[0][0], LDF, 0, D_, wq, 0, acc, wave, lane_lo, hi);
#pragma unroll
  for (int t = 0; t < 3; ++t) {
    int col = (wave + 8*t)*16 + lane_lo;
#pragma unroll
    for (int r = 0; r < 8; ++r)
      qb[(size_t)(8*hi + r)*D_ + col] = acc[t][r] + bq[col];
  }
}

// ---------------------------------------------------------------------------
// K3: attention.  One workgroup per (b, head); softmax over 2048 keys.
// ---------------------------------------------------------------------------
__global__ __launch_bounds__(256) void attn_kernel(
    const float* __restrict__ qb, const float* __restrict__ Kb,
    const float* __restrict__ Vb, float* __restrict__ ao) {
  __shared__ float qv[48];
  __shared__ float att[NN];
  __shared__ float red[256];
  int tid = threadIdx.x;
  int b = blockIdx.x >> 3, h = blockIdx.x & 7;
  if (tid < 48) qv[tid] = qb[(size_t)b*D_ + h*48 + tid];
  __syncthreads();

  const float scale = 0.14433756729740643f;   // 1/sqrt(48)
  float lmax = -1e30f;
  for (int n = tid; n < NN; n += 256) {
    const float* kr = Kb + ((size_t)b*NN + n)*D_ + h*48;
    float d = 0.f;
#pragma unroll
    for (int j = 0; j < 48; ++j) d += qv[j]*kr[j];
    d *= scale;
    att[n] = d;
    lmax = fmaxf(lmax, d);
  }
  red[tid] = lmax;
  __syncthreads();
  for (int s = 128; s >= 1; s >>= 1) {
    if (tid < s) red[tid] = fmaxf(red[tid], red[tid + s]);
    __syncthreads();
  }
  float gmax = red[0];
  __syncthreads();
  float lsum = 0.f;
  for (int n = tid; n < NN; n += 256) {
    float e = expf(att[n] - gmax);
    att[n] = e;
    lsum += e;
  }
  red[tid] = lsum;
  __syncthreads();
  for (int s = 128; s >= 1; s >>= 1) {
    if (tid < s) red[tid] += red[tid + s];
    __syncthreads();
  }
  float inv = 1.0f / red[0];
  __syncthreads();

  int wave = tid >> 5, lane = tid & 31;
  for (int dd = 0; dd < 6; ++dd) {
    int d = wave*6 + dd;                       // 8 waves x 6 dims = 48
    float s = 0.f;
    for (int n = lane; n < NN; n += 32)
      s += att[n] * Vb[((size_t)b*NN + n)*D_ + h*48 + d];
#pragma unroll
    for (int m = 16; m >= 1; m >>= 1) s += __shfl_xor(s, m, 32);
    if (lane == 0) ao[(size_t)b*D_ + h*48 + d] = s * inv;
  }
}

// ---------------------------------------------------------------------------
// K4: fused_q = attout@out_w + out_b + image_embs ; out = fused_q@fc_w + fc_b
// ---------------------------------------------------------------------------
__global__ __launch_bounds__(256) void final_kernel(
    const float* __restrict__ ao, const float* __restrict__ ow,
    const float* __restrict__ ob, const float* __restrict__ embs,
    const float* __restrict__ fcw, const float* __restrict__ fcb,
    float* __restrict__ out) {
  __shared__ float A[16][LDF];
  __shared__ float Fq[16][LDF];
  int tid = threadIdx.x;
  for (int e = tid; e < 16*D_; e += 256) A[e / D_][e % D_] = ao[e];
  __syncthreads();
  int wave = tid >> 5, lane = tid & 31, lane_lo = lane & 15, hi = lane >> 4;
  v8f acc[3] = {};
  gemm_acc3(&A[0][0], LDF, 0, D_, ow, 0, acc, wave, lane_lo, hi);
#pragma unroll
  for (int t = 0; t < 3; ++t) {
    int col = (wave + 8*t)*16 + lane_lo;
#pragma unroll
    for (int r = 0; r < 8; ++r) {
      int m = 8*hi + r;
      Fq[m][col] = acc[t][r] + ob[col] + embs[(size_t)m*D_ + col];
    }
  }
  __syncthreads();
  if (tid < 96) {
    int m = tid / 6, c = tid % 6;
    float s = fcb[c];
    for (int k = 0; k < D_; ++k) s += Fq[m][k] * fcw[k*6 + c];
    out[tid] = s;
  }
}

// ---------------------------------------------------------------------------
// Launch.  Input order (setup_inputs dict, params flattened in-order):
//  0 im0, 1 im1, 2 K, 3 kpts0, 4 kpts1, 5 scores0, 6 scores1, 7 image_embs,
//  8 patch_proj_w, 9 patch_proj_b, 10 ln1_g, 11 ln1_b, 12 pos_w1, 13 pos_b1,
// 14 pos_w2, 15 pos_b2, 16 score_w, 17 score_b, 18 ln2_g, 19 ln2_b,
// 20 wq, 21 bq, 22 wk, 23 bk, 24 wv, 25 bv, 26 out_w, 27 out_b,
// 28 fc_w, 29 fc_b, 30 sorted_matches
// ---------------------------------------------------------------------------
extern "C" void kernel_launch(void* const* d_in, const int* in_sizes, int n_in,
                              void* d_out, int out_size, void* d_ws, size_t ws_size,
                              hipStream_t stream) {
  const float* im0     = (const float*)d_in[0];
  const float* im1     = (const float*)d_in[1];
  const float* Kmat    = (const float*)d_in[2];
  const float* kpts0   = (const float*)d_in[3];
  const float* kpts1   = (const float*)d_in[4];
  const float* scores0 = (const float*)d_in[5];
  const float* scores1 = (const float*)d_in[6];
  const float* embs    = (const float*)d_in[7];
  const float* ppw     = (const float*)d_in[8];
  const float* ppb     = (const float*)d_in[9];
  const float* ln1g    = (const float*)d_in[10];
  const float* ln1b    = (const float*)d_in[11];
  const float* pw1     = (const float*)d_in[12];
  const float* pb1     = (const float*)d_in[13];
  const float* pw2     = (const float*)d_in[14];
  const float* pb2     = (const float*)d_in[15];
  const float* scw     = (const float*)d_in[16];
  const float* scb     = (const float*)d_in[17];
  const float* ln2g    = (const float*)d_in[18];
  const float* ln2b    = (const float*)d_in[19];
  const float* wq      = (const float*)d_in[20];
  const float* bq      = (const float*)d_in[21];
  const float* wk      = (const float*)d_in[22];
  const float* bk      = (const float*)d_in[23];
  const float* wv      = (const float*)d_in[24];
  const float* bv      = (const float*)d_in[25];
  const float* ow      = (const float*)d_in[26];
  const float* ob      = (const float*)d_in[27];
  const float* fcw     = (const float*)d_in[28];
  const float* fcb     = (const float*)d_in[29];
  const int*   matches = (const int*)d_in[30];

  // Workspace layout (floats): coords | scores | q | attout | K | V  (~101 MB)
  float* ws     = (float*)d_ws;
  float* coords = ws;
  float* sc     = coords + (size_t)BN*2;
  float* qb     = sc + BN;
  float* ao     = qb + 16*D_;
  float* Kb     = ao + 16*D_;
  float* Vb     = Kb + (size_t)BN*D_;

  prep_kernel<<<BN/256, 256, 0, stream>>>(matches, kpts0, kpts1,
                                          scores0, scores1, coords, sc);
  fused_patch_kernel<<<BN/16, 256, 0, stream>>>(
      im0, im1, Kmat, ppw, ppb, ln1g, ln1b, pw1, pb1, pw2, pb2,
      scw, scb, ln2g, ln2b, wk, bk, wv, bv, coords, sc, Kb, Vb);
  qproj_kernel<<<1, 256, 0, stream>>>(embs, wq, bq, qb);
  attn_kernel<<<BB*8, 256, 0, stream>>>(qb, Kb, Vb, ao);
  final_kernel<<<1, 256, 0, stream>>>(ao, ow, ob, embs, fcw, fcb, (float*)d_out);
}